// ConstGraphConv_3676492005526
// MI455X (gfx1250) — compile-verified
//
#include <hip/hip_runtime.h>

#define F_IN  128
#define C_OUT 128
#define EPW   256   // edges per wave in the SpMM kernel

typedef __attribute__((ext_vector_type(2))) float v2f;
typedef __attribute__((ext_vector_type(8))) float v8f;

// ---------------------------------------------------------------------------
// Kernel 1: zero the output accumulator (d_out is reused as the fp32 segment-
// sum accumulator, so it must be zeroed deterministically every launch).
// ---------------------------------------------------------------------------
__global__ void zero_kernel(float* __restrict__ out, int n) {
    int i = blockIdx.x * blockDim.x + threadIdx.x;
    if (i < n) out[i] = 0.0f;
}

// ---------------------------------------------------------------------------
// Kernel 2: h = X @ W with V_WMMA_F32_16X16X4_F32 (fp32 exact).
// One wave computes a 16x128 tile of h: the A fragment depends only on k, so
// it is loaded once per k-step and reused across all 8 N-tiles.
//
// Fragment layouts per CDNA5 ISA 7.12.2 (wave32):
//   A 16x4 : lanes 0-15 -> M=lane, VGPR0=K0,VGPR1=K1 ; lanes 16-31 -> K2,K3
//   B 4x16 : lanes 0-15 -> N=lane, VGPR0=K0,VGPR1=K1 ; lanes 16-31 -> K2,K3
//   C/D    : VGPR v -> M = v + 8*(lane>=16), N = lane&15
// ---------------------------------------------------------------------------
__global__ void __launch_bounds__(256)
gemm_wmma_f32(const float* __restrict__ X, const float* __restrict__ W,
              float* __restrict__ H, int n_rows) {
    const int lane = threadIdx.x & 31;
    const int wid  = threadIdx.x >> 5;           // wave id in block (wave32)
    const int half = lane >> 4;                  // 0: lanes 0-15, 1: lanes 16-31
    const int l16  = lane & 15;
    const int m0   = (blockIdx.x * 8 + wid) * 16;
    if (m0 >= n_rows) return;                    // wave-uniform: EXEC stays all-1s

    v8f zero = {};
    v8f acc[8];
#pragma unroll
    for (int t = 0; t < 8; ++t) acc[t] = zero;

    const float* arow = X + (size_t)(m0 + l16) * F_IN;

    for (int k = 0; k < F_IN; k += 4) {
        const int kk = k + 2 * half;
        v2f a;
        a.x = arow[kk];
        a.y = arow[kk + 1];
#pragma unroll
        for (int t = 0; t < 8; ++t) {
            const int n = t * 16 + l16;
            v2f b;
            b.x = W[(size_t)kk * C_OUT + n];
            b.y = W[(size_t)(kk + 1) * C_OUT + n];
            acc[t] = __builtin_amdgcn_wmma_f32_16x16x4_f32(
                /*neg_a=*/false, a, /*neg_b=*/false, b,
                /*c_mod=*/(short)0, acc[t],
                /*reuse_a=*/false, /*reuse_b=*/false);
        }
    }

#pragma unroll
    for (int t = 0; t < 8; ++t) {
        const int n = t * 16 + l16;
#pragma unroll
        for (int v = 0; v < 8; ++v) {
            const int row = m0 + v + 8 * half;
            H[(size_t)row * C_OUT + n] = acc[t][v];
        }
    }
}

// ---------------------------------------------------------------------------
// Kernel 3: SpMM segment-sum over sorted COO rows.
// Each wave owns a contiguous chunk of EPW edges; each lane owns 4 channels.
// Consecutive same-row edges accumulate in registers; flush via f32 atomics
// only on row transitions (rows may span chunk boundaries). h rows are read
// as coalesced float4 (512B/wave) and stay resident in the 192MB L2.
// ---------------------------------------------------------------------------
__global__ void __launch_bounds__(256)
spmm_kernel(const float* __restrict__ H, const float* __restrict__ ev,
            const int* __restrict__ er, const int* __restrict__ ec,
            float* __restrict__ out, int E) {
    const int lane = threadIdx.x & 31;
    const long wave = (long)((blockIdx.x * blockDim.x + threadIdx.x) >> 5);
    long e0 = wave * EPW;
    if (e0 >= E) return;                          // wave-uniform exit
    long e1 = e0 + EPW;
    if (e1 > E) e1 = E;

    // prefetch edge streams ahead into cache (global_prefetch_b8)
    __builtin_prefetch(er + e0 + 64, 0, 1);
    __builtin_prefetch(ec + e0 + 64, 0, 1);
    __builtin_prefetch(ev + e0 + 64, 0, 1);

    const int cb = lane * 4;                      // 4 channels per lane
    float a0 = 0.f, a1 = 0.f, a2 = 0.f, a3 = 0.f;
    int cur = er[e0];

    for (long e = e0; e < e1; ++e) {
        const int r = er[e];
        if (r != cur) {
            float* o = out + (size_t)cur * C_OUT + cb;
            atomicAdd(o + 0, a0);
            atomicAdd(o + 1, a1);
            atomicAdd(o + 2, a2);
            atomicAdd(o + 3, a3);
            a0 = a1 = a2 = a3 = 0.f;
            cur = r;
        }
        const float v = ev[e];
        const long  c = (long)ec[e];
        const float4 hv = *(const float4*)(H + c * C_OUT + cb);
        a0 += v * hv.x;
        a1 += v * hv.y;
        a2 += v * hv.z;
        a3 += v * hv.w;
    }
    float* o = out + (size_t)cur * C_OUT + cb;
    atomicAdd(o + 0, a0);
    atomicAdd(o + 1, a1);
    atomicAdd(o + 2, a2);
    atomicAdd(o + 3, a3);
}

// ---------------------------------------------------------------------------
// Kernel 4: out = relu(out + bias), in place.
// ---------------------------------------------------------------------------
__global__ void bias_relu_kernel(float* __restrict__ out,
                                 const float* __restrict__ bias, int total) {
    int i = blockIdx.x * blockDim.x + threadIdx.x;
    if (i < total) {
        float x = out[i] + bias[i & (C_OUT - 1)];
        out[i] = x > 0.0f ? x : 0.0f;
    }
}

extern "C" void kernel_launch(void* const* d_in, const int* in_sizes, int n_in,
                              void* d_out, int out_size, void* d_ws, size_t ws_size,
                              hipStream_t stream) {
    const float* features = (const float*)d_in[0];   // [N, 128]
    const float* kern     = (const float*)d_in[1];   // [128, 128]
    const float* bias     = (const float*)d_in[2];   // [128]
    const float* ev       = (const float*)d_in[3];   // [E]
    const int*   er       = (const int*)d_in[4];     // [E] sorted rows
    const int*   ec       = (const int*)d_in[5];     // [E] cols

    float* out = (float*)d_out;                      // [N, 128] fp32
    float* H   = (float*)d_ws;                       // [N, 128] scratch for X@W

    const int n_rows = in_sizes[0] / F_IN;
    const int E      = in_sizes[3];
    const int total  = n_rows * C_OUT;

    // 1) zero accumulator
    zero_kernel<<<(total + 255) / 256, 256, 0, stream>>>(out, total);

    // 2) dense transform via fp32 WMMA: 8 waves/block, one 16-row tile/wave
    const int m_tiles = (n_rows + 15) / 16;
    const int gblocks = (m_tiles + 7) / 8;
    gemm_wmma_f32<<<gblocks, 256, 0, stream>>>(features, kern, H, n_rows);

    // 3) SpMM segment-sum (sorted rows, register accumulation, atomic flush)
    const int waves   = (E + EPW - 1) / EPW;
    const int sblocks = (waves + 7) / 8;             // 8 waves per 256-thread block
    spmm_kernel<<<sblocks, 256, 0, stream>>>(H, ev, er, ec, out, E);

    // 4) bias + relu
    bias_relu_kernel<<<(total + 255) / 256, 256, 0, stream>>>(out, bias, total);
}